// Llama4TextExperts_9483287789704
// MI455X (gfx1250) — compile-verified
//
#include <hip/hip_runtime.h>
#include <hip/hip_bf16.h>

// MoE expert FFN (Llama4-style), fp32 end-to-end using V_WMMA_F32_16X16X4_F32.
// Memory-bound (~1.2 GB HBM traffic @ 23.3 TB/s ~ 50us); fp32 WMMA preserves
// reference precision and matrix throughput is far above the roofline need.
// Data movement uses the CDNA5 async global->LDS path (GLOBAL_LOAD_ASYNC_TO_
// LDS_B128, ASYNCcnt) with double-buffered LDS tiles: loads for K-tile i+1 are
// in flight while K-tile i runs on the WMMA pipe; one barrier per K-tile.

typedef float v2f __attribute__((ext_vector_type(2)));
typedef float v8f __attribute__((ext_vector_type(8)));

#define N_EXPERTS  8
#define HIDDEN     2048
#define EXPERT_DIM 4096
#define TOKENS     1024   // tokens per expert

__device__ __forceinline__ v8f wmma_f32(v2f a, v2f b, v8f c) {
    // 8 args: (neg_a, A, neg_b, B, c_mod, C, reuse_a, reuse_b)
    return __builtin_amdgcn_wmma_f32_16x16x4_f32(
        false, a, false, b, (short)0, c, false, false);
}

// Async copy 16B global -> LDS, per-lane LDS destination. Tracked by ASYNCcnt.
__device__ __forceinline__ void async_b128(void* lds, const void* gptr) {
    const unsigned l = (unsigned)(uintptr_t)lds;   // low 32 bits = LDS byte offset
    asm volatile("global_load_async_to_lds_b128 %0, %1, off"
                 :: "v"(l), "v"(gptr) : "memory");
}
__device__ __forceinline__ void wait_async0() {
    asm volatile("s_wait_asynccnt 0x0" ::: "memory");
}

// ---------------------------------------------------------------------------
// Kernel 1: gated[e,t,i] = up * silu(gate), gate/up = X[e] @ W1[e][:, i / i+4096]
// Block tile: 64 tokens x 64 gated columns, 8 wave32s; each wave owns a 16x32
// slab and carries gate+up accumulators (4 x v8f).
// ---------------------------------------------------------------------------
__global__ __launch_bounds__(256) void moe_gateup_silu_kernel(
    const float* __restrict__ X,   // [E*T, H]
    const float* __restrict__ W1,  // [E, H, 2*I]
    float* __restrict__ G)         // [E, T, I]  (workspace)
{
    const int e  = blockIdx.z;
    const int bm = blockIdx.y * 64;          // token tile base
    const int bn = blockIdx.x * 64;          // gated column tile base

    __shared__ float Xs [2][64][20];         // [buf][m][k]  (pad 4 -> bank-safe)
    __shared__ float Wgs[2][16][72];         // [buf][k][n]  (row-major, pad 8)
    __shared__ float Wus[2][16][72];

    const int tid  = threadIdx.x;
    const int lane = tid & 31;
    const int wave = tid >> 5;
    const int mt   = wave >> 1;              // 0..3 : M tile (16 rows)
    const int nt0  = (wave & 1) * 2;         // 0/2  : first of two N tiles

    const float* Xg = X  + (size_t)(e * TOKENS + bm) * HIDDEN;
    const float* Wg = W1 + (size_t)e * HIDDEN * (2 * EXPERT_DIM);

    v8f accg0 = {}, accg1 = {}, accu0 = {}, accu1 = {};

    // cooperative async-load coordinates (256 threads x 16B per tile)
    const int xrow = tid >> 2;               // 0..63
    const int xcol = (tid & 3) * 4;          // 0..12
    const int wk   = tid >> 4;               // 0..15
    const int wn   = (tid & 15) * 4;         // 0..60

    // prologue: stage K-tile 0 into buffer 0
    async_b128(&Xs [0][xrow][xcol], Xg + (size_t)xrow * HIDDEN + xcol);
    async_b128(&Wgs[0][wk][wn], Wg + (size_t)wk * (2 * EXPERT_DIM) + bn + wn);
    async_b128(&Wus[0][wk][wn], Wg + (size_t)wk * (2 * EXPERT_DIM) + EXPERT_DIM + bn + wn);

    for (int k0 = 0; k0 < HIDDEN; k0 += 16) {
        const int cur = (k0 >> 4) & 1;
        wait_async0();                       // my cur-tile bytes are in LDS
        __syncthreads();                     // everyone's are; nxt is readable->writable

        if (k0 + 16 < HIDDEN) {              // launch next K-tile into other buffer
            const int nxt = cur ^ 1;
            const int kn  = k0 + 16;
            async_b128(&Xs [nxt][xrow][xcol], Xg + (size_t)xrow * HIDDEN + kn + xcol);
            async_b128(&Wgs[nxt][wk][wn],
                       Wg + (size_t)(kn + wk) * (2 * EXPERT_DIM) + bn + wn);
            async_b128(&Wus[nxt][wk][wn],
                       Wg + (size_t)(kn + wk) * (2 * EXPERT_DIM) + EXPERT_DIM + bn + wn);
        }

        #pragma unroll
        for (int kk = 0; kk < 16; kk += 4) {
            const int klo = kk + ((lane >> 4) << 1);   // lanes16-31 -> K+2
            const int ar  = mt * 16 + (lane & 15);
            v2f a;   a.x   = Xs[cur][ar][klo];   a.y   = Xs[cur][ar][klo + 1];
            const int nr0 = nt0 * 16 + (lane & 15);
            const int nr1 = nr0 + 16;
            v2f bg0; bg0.x = Wgs[cur][klo][nr0]; bg0.y = Wgs[cur][klo + 1][nr0];
            v2f bg1; bg1.x = Wgs[cur][klo][nr1]; bg1.y = Wgs[cur][klo + 1][nr1];
            v2f bu0; bu0.x = Wus[cur][klo][nr0]; bu0.y = Wus[cur][klo + 1][nr0];
            v2f bu1; bu1.x = Wus[cur][klo][nr1]; bu1.y = Wus[cur][klo + 1][nr1];
            accg0 = wmma_f32(a, bg0, accg0);
            accg1 = wmma_f32(a, bg1, accg1);
            accu0 = wmma_f32(a, bu0, accu0);
            accu1 = wmma_f32(a, bu1, accu1);
        }
    }

    // epilogue: gated = up * silu(gate), straight to workspace.
    // C/D layout: VGPR r -> M=r (lanes 0-15) / M=8+r (lanes 16-31), N=lane%16
    const int rbase = bm + mt * 16 + ((lane >> 4) << 3);
    const int cbase = bn + nt0 * 16 + (lane & 15);
    float* Gout = G + (size_t)e * TOKENS * EXPERT_DIM;
    #pragma unroll
    for (int r = 0; r < 8; ++r) {
        const float g0 = accg0[r], u0 = accu0[r];
        const float g1 = accg1[r], u1 = accu1[r];
        const float s0 = u0 * (g0 / (1.0f + __expf(-g0)));
        const float s1 = u1 * (g1 / (1.0f + __expf(-g1)));
        float* row = Gout + (size_t)(rbase + r) * EXPERT_DIM + cbase;
        row[0]  = s0;
        row[16] = s1;
    }
}

// ---------------------------------------------------------------------------
// Kernel 2: out[e,t,h] = gated[e] @ W2[e]   (K = 4096)
// ---------------------------------------------------------------------------
__global__ __launch_bounds__(256) void moe_down_kernel(
    const float* __restrict__ G,   // [E, T, I]
    const float* __restrict__ W2,  // [E, I, H]
    float* __restrict__ Out)       // [E*T, H]
{
    const int e  = blockIdx.z;
    const int bm = blockIdx.y * 64;
    const int bn = blockIdx.x * 64;

    __shared__ float Gs[2][64][20];   // [buf][m][k]
    __shared__ float Ws[2][16][72];   // [buf][k][n]

    const int tid  = threadIdx.x;
    const int lane = tid & 31;
    const int wave = tid >> 5;
    const int mt   = wave >> 1;
    const int nt0  = (wave & 1) * 2;

    const float* Gg = G  + ((size_t)e * TOKENS + bm) * EXPERT_DIM;
    const float* Wd = W2 + (size_t)e * EXPERT_DIM * HIDDEN;

    v8f acc0 = {}, acc1 = {};

    const int grow = tid >> 2;
    const int gcol = (tid & 3) * 4;
    const int wk   = tid >> 4;
    const int wn   = (tid & 15) * 4;

    async_b128(&Gs[0][grow][gcol], Gg + (size_t)grow * EXPERT_DIM + gcol);
    async_b128(&Ws[0][wk][wn],     Wd + (size_t)wk * HIDDEN + bn + wn);

    for (int k0 = 0; k0 < EXPERT_DIM; k0 += 16) {
        const int cur = (k0 >> 4) & 1;
        wait_async0();
        __syncthreads();

        if (k0 + 16 < EXPERT_DIM) {
            const int nxt = cur ^ 1;
            const int kn  = k0 + 16;
            async_b128(&Gs[nxt][grow][gcol], Gg + (size_t)grow * EXPERT_DIM + kn + gcol);
            async_b128(&Ws[nxt][wk][wn],     Wd + (size_t)(kn + wk) * HIDDEN + bn + wn);
        }

        #pragma unroll
        for (int kk = 0; kk < 16; kk += 4) {
            const int klo = kk + ((lane >> 4) << 1);
            const int ar  = mt * 16 + (lane & 15);
            v2f a;  a.x  = Gs[cur][ar][klo];   a.y  = Gs[cur][ar][klo + 1];
            const int nr0 = nt0 * 16 + (lane & 15);
            const int nr1 = nr0 + 16;
            v2f b0; b0.x = Ws[cur][klo][nr0];  b0.y = Ws[cur][klo + 1][nr0];
            v2f b1; b1.x = Ws[cur][klo][nr1];  b1.y = Ws[cur][klo + 1][nr1];
            acc0 = wmma_f32(a, b0, acc0);
            acc1 = wmma_f32(a, b1, acc1);
        }
    }

    const int rbase = bm + mt * 16 + ((lane >> 4) << 3);
    const int cbase = bn + nt0 * 16 + (lane & 15);
    float* O = Out + (size_t)e * TOKENS * HIDDEN;
    #pragma unroll
    for (int r = 0; r < 8; ++r) {
        float* row = O + (size_t)(rbase + r) * HIDDEN + cbase;
        row[0]  = acc0[r];
        row[16] = acc1[r];
    }
}

extern "C" void kernel_launch(void* const* d_in, const int* in_sizes, int n_in,
                              void* d_out, int out_size, void* d_ws, size_t ws_size,
                              hipStream_t stream) {
    const float* X  = (const float*)d_in[0];   // hidden_states [E*T, H]
    const float* W1 = (const float*)d_in[1];   // gate_up_proj  [E, H, 2I]
    const float* W2 = (const float*)d_in[2];   // down_proj     [E, I, H]
    float* out   = (float*)d_out;              // [E*T, H]
    float* gated = (float*)d_ws;               // needs E*T*I*4 = 128 MiB

    dim3 g1(EXPERT_DIM / 64, TOKENS / 64, N_EXPERTS);   // 64 x 16 x 8
    moe_gateup_silu_kernel<<<g1, 256, 0, stream>>>(X, W1, gated);

    dim3 g2(HIDDEN / 64, TOKENS / 64, N_EXPERTS);       // 32 x 16 x 8
    moe_down_kernel<<<g2, 256, 0, stream>>>(gated, W2, out);
}